// HAN_8134668058629
// MI455X (gfx1250) — compile-verified
//
#include <hip/hip_runtime.h>

typedef float v2f __attribute__((ext_vector_type(2)));
typedef float v4f __attribute__((ext_vector_type(4)));
typedef float v8f __attribute__((ext_vector_type(8)));

#define SLOPE 0.01f

__device__ __forceinline__ float leaky(float x) { return x > 0.f ? x : SLOPE * x; }

// monotone float -> uint key for atomicMax-based segment max
__device__ __forceinline__ unsigned fkey(float f) {
  unsigned u = __float_as_uint(f);
  return (u & 0x80000000u) ? ~u : (u | 0x80000000u);
}
__device__ __forceinline__ float fdec(unsigned k) {
  return __uint_as_float((k & 0x80000000u) ? (k ^ 0x80000000u) : ~k);
}

__device__ __forceinline__ void atomicAddF(float* p, float v) {
  unsafeAtomicAdd(p, v); // native global_atomic_add_f32
}

// ---------------------------------------------------------------------------
// Stage A: node feature projection  C[M,64] = A[M,K] @ Bw[K,64] + bias
// One wave per 16x64 output stripe: 4 col-tiles share each A b64 load.
// fp32 WMMA 16x16x4; M-tail handled by row clamp (stores guarded).
// ---------------------------------------------------------------------------
__global__ void proj_gemm_kernel(const float* __restrict__ A, const float* __restrict__ Bw,
                                 const float* __restrict__ bias, float* __restrict__ C,
                                 int M, int K) {
  const int N = 64;
  const int l = threadIdx.x;
  const int lh = l & 15;
  const int hi = (l >> 4) & 1;
  const int m0 = blockIdx.x * 16;
  const int row = m0 + lh;
  const int rowc = row < M ? row : (M - 1);      // clamp: branch-free loads

  const float* ap = A + (size_t)rowc * K + (hi ? 2 : 0);
  const float* bp = Bw + (size_t)(hi ? 2 : 0) * N + lh;

  v8f c0 = {}, c1 = {}, c2 = {}, c3 = {};
  for (int k = 0; k < K; k += 4) {
    v2f a = *(const v2f*)ap;  ap += 4;
    v2f b;
    b.x = bp[0];       b.y = bp[N];
    c0 = __builtin_amdgcn_wmma_f32_16x16x4_f32(false, a, false, b, (short)0, c0, false, false);
    b.x = bp[16];      b.y = bp[16 + N];
    c1 = __builtin_amdgcn_wmma_f32_16x16x4_f32(false, a, false, b, (short)0, c1, false, false);
    b.x = bp[32];      b.y = bp[32 + N];
    c2 = __builtin_amdgcn_wmma_f32_16x16x4_f32(false, a, false, b, (short)0, c2, false, false);
    b.x = bp[48];      b.y = bp[48 + N];
    c3 = __builtin_amdgcn_wmma_f32_16x16x4_f32(false, a, false, b, (short)0, c3, false, false);
    bp += 4 * N;
  }
  const float bi0 = bias[lh], bi1 = bias[16 + lh], bi2 = bias[32 + lh], bi3 = bias[48 + lh];
#pragma unroll
  for (int r = 0; r < 8; ++r) {
    int orow = m0 + r + (hi ? 8 : 0);
    if (orow < M) {
      float* cp = C + (size_t)orow * N + lh;
      cp[0]  = c0[r] + bi0;
      cp[16] = c1[r] + bi1;
      cp[32] = c2[r] + bi2;
      cp[48] = c3[r] + bi3;
    }
  }
}

// ---------------------------------------------------------------------------
// Stage B: per-node attention pre-dots (GAT factorization)
// ---------------------------------------------------------------------------
__global__ void alpha_kernel(const float* __restrict__ nf,
                             const float* __restrict__ att0, const float* __restrict__ att1,
                             float* __restrict__ ad0, float* __restrict__ as0,
                             float* __restrict__ ad1, float* __restrict__ as1, int Nn) {
  int t = blockIdx.x * blockDim.x + threadIdx.x;
  if (t >= Nn * 8) return;
  int n = t >> 3, h = t & 7;
  const float* f = nf + (size_t)n * 64;
  const float* a0 = att0 + h * 128;
  const float* a1 = att1 + h * 128;
  float d0 = 0.f, s0 = 0.f, d1 = 0.f, s1 = 0.f;
#pragma unroll 8
  for (int d = 0; d < 64; ++d) {
    float v = f[d];
    d0 += a0[d] * v;      s0 += a0[64 + d] * v;
    d1 += a1[d] * v;      s1 += a1[64 + d] * v;
  }
  ad0[t] = d0; as0[t] = s0; ad1[t] = d1; as1[t] = s1;
}

// ---------------------------------------------------------------------------
// slot map: slot[target_nodes[b]] = b (duplicates collapse to one winner)
// ---------------------------------------------------------------------------
__global__ void mark_kernel(const int* __restrict__ tn, int* __restrict__ slot, int Bn) {
  int b = blockIdx.x * blockDim.x + threadIdx.x;
  if (b < Bn) slot[tn[b]] = b;
}

// ---------------------------------------------------------------------------
// Stage D pass 1: segment max of leaky(attn logit) over edges hitting targets
// ---------------------------------------------------------------------------
__global__ void edge_max_kernel(const int* __restrict__ e0, const int* __restrict__ e1,
                                const int* __restrict__ slot,
                                const float* __restrict__ ad0, const float* __restrict__ as0,
                                const float* __restrict__ ad1, const float* __restrict__ as1,
                                unsigned* __restrict__ mkeys, int En, int Bn) {
  int e = blockIdx.x * blockDim.x + threadIdx.x;
  if (e >= En) return;
#pragma unroll
  for (int mp = 0; mp < 2; ++mp) {
    const int* ed = mp ? e1 : e0;
    int src = ed[2 * e], dst = ed[2 * e + 1];
    int s = slot[dst];
    if (s < 0) continue;
    const float* ad = mp ? ad1 : ad0;
    const float* as = mp ? as1 : as0;
    unsigned* mk = mkeys + (size_t)(mp * Bn + s) * 8;
#pragma unroll
    for (int h = 0; h < 8; ++h) {
      float ev = leaky(ad[dst * 8 + h] + as[src * 8 + h]);
      atomicMax(&mk[h], fkey(ev));
    }
  }
}

// ---------------------------------------------------------------------------
// Stage D pass 2: one wave per (metapath, edge): accumulate den and ex*hs
// ---------------------------------------------------------------------------
__global__ void edge_agg_kernel(const int* __restrict__ e0, const int* __restrict__ e1,
                                const int* __restrict__ slot, const float* __restrict__ nf,
                                const float* __restrict__ ad0, const float* __restrict__ as0,
                                const float* __restrict__ ad1, const float* __restrict__ as1,
                                const unsigned* __restrict__ mkeys,
                                float* __restrict__ den, float* __restrict__ acc,
                                int En, int Bn) {
  int w = blockIdx.x * (blockDim.x >> 5) + (threadIdx.x >> 5);
  int lane = threadIdx.x & 31;
  if (w >= 2 * En) return;
  int mp = (w >= En) ? 1 : 0;
  int e = mp ? (w - En) : w;
  const int* ed = mp ? e1 : e0;
  int src = ed[2 * e], dst = ed[2 * e + 1];
  int s = slot[dst];
  if (s < 0) return;
  const float* ad = mp ? ad1 : ad0;
  const float* as = mp ? as1 : as0;
  size_t base = (size_t)(mp * Bn + s) * 8;
  float ex[8];
#pragma unroll
  for (int h = 0; h < 8; ++h) {
    float ev = leaky(ad[dst * 8 + h] + as[src * 8 + h]);
    ex[h] = __expf(ev - fdec(mkeys[base + h]));
  }
  if (lane == 0) {
#pragma unroll
    for (int h = 0; h < 8; ++h) atomicAddF(&den[base + h], ex[h]);
  }
  float hs0 = nf[(size_t)src * 64 + lane];
  float hs1 = nf[(size_t)src * 64 + lane + 32];
#pragma unroll
  for (int h = 0; h < 8; ++h) {
    float* ap = acc + (base + h) * 64;
    atomicAddF(&ap[lane], ex[h] * hs0);
    atomicAddF(&ap[lane + 32], ex[h] * hs1);
  }
}

// ---------------------------------------------------------------------------
// Stage D pass 3: normalize, leaky, expand slots -> per-target rows (2B x 512)
// Each lane owns 16 consecutive floats = one quarter of one head -> one den
// load per lane, b128 traffic for acc/hbuf.
// ---------------------------------------------------------------------------
__global__ void norm_expand_kernel(const int* __restrict__ tn, const int* __restrict__ slot,
                                   const float* __restrict__ den, const float* __restrict__ acc,
                                   float* __restrict__ hbuf, int Bn) {
  int w = blockIdx.x * (blockDim.x >> 5) + (threadIdx.x >> 5);
  int lane = threadIdx.x & 31;
  if (w >= 2 * Bn) return;
  int mp = (w >= Bn) ? 1 : 0;
  int b = mp ? (w - Bn) : w;
  int s = slot[tn[b]];
  size_t base = (size_t)(mp * Bn + s) * 8;
  int h = lane >> 2;                 // head owned by this lane
  int d0 = (lane & 3) * 16;          // starting dim within head
  float dv = den[base + h];
  float inv = (dv > 0.f) ? 1.f / dv : 0.f;
  const float* ap = acc + (base + h) * 64 + d0;
  float* out = hbuf + (size_t)(mp * Bn + b) * 512 + h * 64 + d0;
#pragma unroll
  for (int j = 0; j < 4; ++j) {
    v4f v = *(const v4f*)(ap + j * 4);
    v4f o;
#pragma unroll
    for (int i = 0; i < 4; ++i) o[i] = leaky(v[i] * inv);
    *(v4f*)(out + j * 4) = o;
  }
}

// ---------------------------------------------------------------------------
// Stage E: semantic attention GEMM (2B x 512)@(512 x 128), WMMA fp32.
// grid.y in {0,1}: each wave covers 4 col-tiles (64 cols) sharing A loads.
// Epilogue: tanh, dot with a_sem, wave-reduce; only 2 running scalars kept.
// ---------------------------------------------------------------------------
__global__ void sem_gemm_kernel(const float* __restrict__ A, const float* __restrict__ Bw,
                                const float* __restrict__ bsem, const float* __restrict__ a_sem,
                                float* __restrict__ sc_sum, int Bn) {
  const int K = 512, N = 128;
  const int l = threadIdx.x;
  const int lh = l & 15;
  const int hi = (l >> 4) & 1;
  const int m0 = blockIdx.x * 16;
  const int n0 = blockIdx.y * 64;

  const float* ap = A + (size_t)(m0 + lh) * K + (hi ? 2 : 0);
  const float* bp = Bw + (size_t)(hi ? 2 : 0) * N + n0 + lh;

  v8f c0 = {}, c1 = {}, c2 = {}, c3 = {};
  for (int k = 0; k < K; k += 4) {
    v2f a = *(const v2f*)ap;  ap += 4;
    v2f b;
    b.x = bp[0];       b.y = bp[N];
    c0 = __builtin_amdgcn_wmma_f32_16x16x4_f32(false, a, false, b, (short)0, c0, false, false);
    b.x = bp[16];      b.y = bp[16 + N];
    c1 = __builtin_amdgcn_wmma_f32_16x16x4_f32(false, a, false, b, (short)0, c1, false, false);
    b.x = bp[32];      b.y = bp[32 + N];
    c2 = __builtin_amdgcn_wmma_f32_16x16x4_f32(false, a, false, b, (short)0, c2, false, false);
    b.x = bp[48];      b.y = bp[48 + N];
    c3 = __builtin_amdgcn_wmma_f32_16x16x4_f32(false, a, false, b, (short)0, c3, false, false);
    bp += 4 * N;
  }
  float s = 0.f;
  {
    const float bs = bsem[n0 + lh], av = a_sem[n0 + lh];
#pragma unroll
    for (int r = 0; r < 8; ++r) s += av * tanhf(c0[r] + bs);
  }
  {
    const float bs = bsem[n0 + 16 + lh], av = a_sem[n0 + 16 + lh];
#pragma unroll
    for (int r = 0; r < 8; ++r) s += av * tanhf(c1[r] + bs);
  }
  {
    const float bs = bsem[n0 + 32 + lh], av = a_sem[n0 + 32 + lh];
#pragma unroll
    for (int r = 0; r < 8; ++r) s += av * tanhf(c2[r] + bs);
  }
  {
    const float bs = bsem[n0 + 48 + lh], av = a_sem[n0 + 48 + lh];
#pragma unroll
    for (int r = 0; r < 8; ++r) s += av * tanhf(c3[r] + bs);
  }
  for (int off = 16; off > 0; off >>= 1) s += __shfl_xor(s, off, 32);
  if (l == 0) atomicAddF(&sc_sum[(m0 >= Bn) ? 1 : 0], s);
}

__global__ void beta_kernel(const float* __restrict__ sc_sum, float* __restrict__ beta, int Bn) {
  float m0 = sc_sum[0] / (float)Bn, m1 = sc_sum[1] / (float)Bn;
  float mx = fmaxf(m0, m1);
  float z0 = __expf(m0 - mx), z1 = __expf(m1 - mx);
  float inv = 1.f / (z0 + z1);
  beta[0] = z0 * inv;
  beta[1] = z1 * inv;
}

// ---------------------------------------------------------------------------
// Final: embeddings = beta0*h0 + beta1*h1 ; cls = emb @ Wcls + bcls
// One wave per target row; each lane owns 16 consecutive floats (b128 ops);
// 5 classifier dots reduced in-wave.
// ---------------------------------------------------------------------------
__global__ void final_kernel(const float* __restrict__ hbuf, const float* __restrict__ beta,
                             const float* __restrict__ Wcls, const float* __restrict__ bcls,
                             float* __restrict__ out_cls, float* __restrict__ out_emb, int Bn) {
  int w = blockIdx.x * (blockDim.x >> 5) + (threadIdx.x >> 5);
  int lane = threadIdx.x & 31;
  if (w >= Bn) return;
  float b0 = beta[0], b1 = beta[1];
  const float* h0 = hbuf + (size_t)w * 512 + lane * 16;
  const float* h1 = hbuf + (size_t)(Bn + w) * 512 + lane * 16;
  float* emb = out_emb + (size_t)w * 512 + lane * 16;
  float p[5] = {0.f, 0.f, 0.f, 0.f, 0.f};
#pragma unroll
  for (int j = 0; j < 4; ++j) {
    v4f a = *(const v4f*)(h0 + j * 4);
    v4f bq = *(const v4f*)(h1 + j * 4);
    v4f o;
#pragma unroll
    for (int i = 0; i < 4; ++i) {
      float v = b0 * a[i] + b1 * bq[i];
      o[i] = v;
      int idx = lane * 16 + j * 4 + i;
#pragma unroll
      for (int cc = 0; cc < 5; ++cc) p[cc] += v * Wcls[idx * 5 + cc];
    }
    *(v4f*)(emb + j * 4) = o;
  }
#pragma unroll
  for (int cc = 0; cc < 5; ++cc)
    for (int off = 16; off > 0; off >>= 1) p[cc] += __shfl_xor(p[cc], off, 32);
  if (lane == 0) {
#pragma unroll
    for (int cc = 0; cc < 5; ++cc) out_cls[(size_t)w * 5 + cc] = p[cc] + bcls[cc];
  }
}

extern "C" void kernel_launch(void* const* d_in, const int* in_sizes, int n_in,
                              void* d_out, int out_size, void* d_ws, size_t ws_size,
                              hipStream_t stream) {
  (void)n_in; (void)out_size; (void)ws_size;
  const int* tn     = (const int*)d_in[0];
  const int* m0e    = (const int*)d_in[1];
  const int* m1e    = (const int*)d_in[2];
  const float* feat0 = (const float*)d_in[4];
  const float* feat1 = (const float*)d_in[5];
  const float* W0   = (const float*)d_in[6];
  const float* b0   = (const float*)d_in[7];
  const float* W1   = (const float*)d_in[8];
  const float* b1   = (const float*)d_in[9];
  const float* att0 = (const float*)d_in[10];
  const float* att1 = (const float*)d_in[11];
  const float* Wsem = (const float*)d_in[12];
  const float* bsem = (const float*)d_in[13];
  const float* a_sem = (const float*)d_in[14];
  const float* Wcls = (const float*)d_in[15];
  const float* bcls = (const float*)d_in[16];

  const int Bn = in_sizes[0];
  const int En = in_sizes[1] / 2;
  const int Nn = in_sizes[3];
  const int N0 = in_sizes[4] / 128;
  const int N1 = in_sizes[5] / 64;

  char* ws = (char*)d_ws;
  size_t off = 0;
  auto alloc = [&](size_t bytes) -> void* {
    void* p = ws + off;
    off = (off + bytes + 15) & ~(size_t)15;
    return p;
  };
  float* nf   = (float*)alloc((size_t)Nn * 64 * 4);
  float* ad0  = (float*)alloc((size_t)Nn * 8 * 4);
  float* as0  = (float*)alloc((size_t)Nn * 8 * 4);
  float* ad1  = (float*)alloc((size_t)Nn * 8 * 4);
  float* as1  = (float*)alloc((size_t)Nn * 8 * 4);
  int* slot   = (int*)alloc((size_t)Nn * 4);
  char* zbase = ws + off;                       // zero-init region start
  unsigned* mkeys = (unsigned*)alloc((size_t)2 * Bn * 8 * 4);
  float* den  = (float*)alloc((size_t)2 * Bn * 8 * 4);
  float* acc  = (float*)alloc((size_t)2 * Bn * 512 * 4);
  float* sc_sum = (float*)alloc(16);
  size_t zlen = (size_t)((ws + off) - zbase);
  float* hbuf = (float*)alloc((size_t)2 * Bn * 512 * 4);
  float* beta = (float*)alloc(16);

  hipMemsetAsync(slot, 0xFF, (size_t)Nn * 4, stream);   // slot = -1
  hipMemsetAsync(zbase, 0, zlen, stream);               // mkeys/den/acc/sc_sum = 0

  mark_kernel<<<(Bn + 255) / 256, 256, 0, stream>>>(tn, slot, Bn);

  proj_gemm_kernel<<<(N0 + 15) / 16, 32, 0, stream>>>(feat0, W0, b0, nf, N0, 128);
  proj_gemm_kernel<<<(N1 + 15) / 16, 32, 0, stream>>>(feat1, W1, b1,
                                                      nf + (size_t)N0 * 64, N1, 64);

  alpha_kernel<<<(Nn * 8 + 255) / 256, 256, 0, stream>>>(nf, att0, att1, ad0, as0, ad1, as1, Nn);

  edge_max_kernel<<<(En + 255) / 256, 256, 0, stream>>>(m0e, m1e, slot, ad0, as0, ad1, as1,
                                                        mkeys, En, Bn);

  {
    int waves = 2 * En;
    edge_agg_kernel<<<(waves + 7) / 8, 256, 0, stream>>>(m0e, m1e, slot, nf, ad0, as0, ad1, as1,
                                                         mkeys, den, acc, En, Bn);
  }
  {
    int waves = 2 * Bn;
    norm_expand_kernel<<<(waves + 7) / 8, 256, 0, stream>>>(tn, slot, den, acc, hbuf, Bn);
  }

  sem_gemm_kernel<<<dim3((2 * Bn) / 16, 2), 32, 0, stream>>>(hbuf, Wsem, bsem, a_sem, sc_sum, Bn);
  beta_kernel<<<1, 1, 0, stream>>>(sc_sum, beta, Bn);

  float* out_cls = (float*)d_out;
  float* out_emb = out_cls + (size_t)Bn * 5;
  final_kernel<<<(Bn + 7) / 8, 256, 0, stream>>>(hbuf, beta, Wcls, bcls, out_cls, out_emb, Bn);
}